// MultiBoxLoss_71038759076487
// MI455X (gfx1250) — compile-verified
//
#include <hip/hip_runtime.h>
#include <math.h>

#define NB 64
#define NP 8732
#define NO 16
#define NC 21
#define BP (NB * NP)
#define THRESH 0.5f
#define NEGPOS 3
#define TILE_ROWS 256
#define TILE_ELEMS (TILE_ROWS * NC)   // 5376 f32 = 21504 B LDS per block

typedef __attribute__((ext_vector_type(2))) float v2f;
typedef __attribute__((ext_vector_type(8))) float v8f;
typedef __attribute__((ext_vector_type(4))) unsigned int u32x4;
typedef __attribute__((ext_vector_type(8))) int i32x8;
typedef __attribute__((ext_vector_type(4))) int i32x4;

// ---- exact fp32 wave32 reduction via V_WMMA_F32_16X16X4_F32 ----
// A: 16x4, lane L(0-15)->(M=L,K=0), lane L(16-31)->(M=L-16,K=2) in VGPR0; VGPR1 zero.
// B: 4x16 all-ones => D[m,n] = rowsum(m) = partial[m] + partial[m+16].
// Sum of D's 8 VGPRs per lane: lanes 0-15 = sum(rows 0..7), lanes 16-31 = sum(rows 8..15).
__device__ __forceinline__ float wave_reduce_f32_wmma(float partial) {
  v2f a; a[0] = partial; a[1] = 0.0f;
  v2f bm; bm[0] = 1.0f; bm[1] = 1.0f;
  v8f c = {0.f, 0.f, 0.f, 0.f, 0.f, 0.f, 0.f, 0.f};
  c = __builtin_amdgcn_wmma_f32_16x16x4_f32(false, a, false, bm, (short)0, c, false, false);
  float s = c[0] + c[1] + c[2] + c[3] + c[4] + c[5] + c[6] + c[7];
  s += __shfl_xor(s, 16, 32);
  return s;  // every lane holds the full 32-lane sum
}

// ---------------- Kernel 1: match + encode (one block per batch) ----------------
__global__ void k_match(const float* __restrict__ priors,
                        const float* __restrict__ targets,
                        float* __restrict__ bto, int* __restrict__ bti,
                        float* __restrict__ loc_t, int* __restrict__ conf_t,
                        int* __restrict__ num_pos, int* __restrict__ npos_total) {
  const int b = blockIdx.x;
  const int tid = threadIdx.x;
  __shared__ float tr[NO][5];     // xyxy + label
  __shared__ float areaT[NO];
  __shared__ float red_v[NO][256];
  __shared__ int   red_i[NO][256];
  __shared__ int   bpidx[NO];
  __shared__ int   s_pos;

  if (tid < NO * 5) ((float*)tr)[tid] = targets[b * NO * 5 + tid];
  if (tid == 0) s_pos = 0;
  __syncthreads();
  if (tid < NO) areaT[tid] = (tr[tid][2] - tr[tid][0]) * (tr[tid][3] - tr[tid][1]);
  __syncthreads();

  float tbv[NO]; int tbp[NO];
#pragma unroll
  for (int o = 0; o < NO; ++o) { tbv[o] = -1.0f; tbp[o] = 0; }

  for (int p = tid; p < NP; p += 256) {
    float pcx = priors[p * 4 + 0], pcy = priors[p * 4 + 1];
    float pw  = priors[p * 4 + 2], ph  = priors[p * 4 + 3];
    float px0 = pcx - pw * 0.5f, py0 = pcy - ph * 0.5f;
    float px1 = pcx + pw * 0.5f, py1 = pcy + ph * 0.5f;
    float areaP = (px1 - px0) * (py1 - py0);
    float bv = -1.0f; int bo = 0;
#pragma unroll
    for (int o = 0; o < NO; ++o) {
      float ix0 = fmaxf(tr[o][0], px0), iy0 = fmaxf(tr[o][1], py0);
      float ix1 = fminf(tr[o][2], px1), iy1 = fminf(tr[o][3], py1);
      float iw = fmaxf(ix1 - ix0, 0.0f), ih = fmaxf(iy1 - iy0, 0.0f);
      float inter = iw * ih;
      float iou = inter / (areaT[o] + areaP - inter);
      if (iou > bv) { bv = iou; bo = o; }          // first-max over truths
      if (iou > tbv[o]) { tbv[o] = iou; tbp[o] = p; }  // first-max within stride
    }
    bto[b * NP + p] = bv;
    bti[b * NP + p] = bo;
  }
#pragma unroll
  for (int o = 0; o < NO; ++o) { red_v[o][tid] = tbv[o]; red_i[o][tid] = tbp[o]; }
  __syncthreads();
  if (tid < NO) {  // best prior per truth, ties -> smallest prior index
    float bv = -1.0f; int bp = 0x7fffffff;
    for (int t = 0; t < 256; ++t) {
      float v = red_v[tid][t]; int i = red_i[tid][t];
      if (v > bv || (v == bv && i < bp)) { bv = v; bp = i; }
    }
    bpidx[tid] = bp;
  }
  __syncthreads();
  if (tid == 0) {  // sequential scatter: later truth wins on duplicate prior
    for (int o = 0; o < NO; ++o) {
      int pp = bpidx[o];
      bto[b * NP + pp] = 2.0f;
      bti[b * NP + pp] = o;
    }
  }
  __threadfence();
  __syncthreads();

  int lpos = 0;
  for (int p = tid; p < NP; p += 256) {
    float ov = bto[b * NP + p];
    int ti = bti[b * NP + p];
    int c = (ov < THRESH) ? 0 : (int)(tr[ti][4] + 1.0f);
    conf_t[b * NP + p] = c;
    if (c > 0) ++lpos;
    float pcx = priors[p * 4 + 0], pcy = priors[p * 4 + 1];
    float pw  = priors[p * 4 + 2], ph  = priors[p * 4 + 3];
    float m0 = tr[ti][0], m1 = tr[ti][1], m2 = tr[ti][2], m3 = tr[ti][3];
    int base = (b * NP + p) * 4;
    loc_t[base + 0] = ((m0 + m2) * 0.5f - pcx) / (0.1f * pw);
    loc_t[base + 1] = ((m1 + m3) * 0.5f - pcy) / (0.1f * ph);
    loc_t[base + 2] = logf(fmaxf(m2 - m0, 1e-8f) / pw) / 0.2f;
    loc_t[base + 3] = logf(fmaxf(m3 - m1, 1e-8f) / ph) / 0.2f;
  }
  atomicAdd(&s_pos, lpos);
  __syncthreads();
  if (tid == 0) { num_pos[b] = s_pos; atomicAdd(npos_total, s_pos); }
}

// ------- Kernel 2: TDM-staged per-row CE + smooth-L1 with WMMA wave reduction -------
__global__ void k_confrow(const float* __restrict__ conf_data,
                          const float* __restrict__ loc_data,
                          const float* __restrict__ loc_t,
                          const int* __restrict__ conf_t,
                          float* __restrict__ ce,
                          float* __restrict__ acc /* [0]=loss_l */) {
  __shared__ float sconf[TILE_ELEMS];
  const int tid = threadIdx.x;
  const long long start = (long long)blockIdx.x * TILE_ELEMS;
  const long long total = (long long)BP * NC;

  if (tid < 32) {  // wave 0 issues one TDM copy for the whole block's tile
    unsigned long long gaddr =
        (unsigned long long)(uintptr_t)conf_data + (unsigned long long)start * 4ull;
    unsigned int ldsoff = (unsigned int)(uintptr_t)&sconf[0];  // low 32 bits = LDS byte offset
    unsigned long long t0 = (unsigned long long)(total - start);  // remaining elems (OOB reads -> 0)

    u32x4 g0;
    g0[0] = 1u;                                   // count=1, gather off
    g0[1] = ldsoff;                               // lds_addr
    g0[2] = (unsigned int)(gaddr & 0xffffffffull);
    g0[3] = (unsigned int)((gaddr >> 32) & 0x1ffffffull) | (2u << 30);  // type=2 (image)

    i32x8 g1;
    g1[0] = (int)(2u << 16);                                   // data_size=4B
    g1[1] = (int)((unsigned int)(t0 & 0xffffull) << 16);       // tensor_dim0[15:0]
    g1[2] = (int)((unsigned int)((t0 >> 16) & 0xffffull) | (1u << 16));  // dim0 hi | tensor_dim1=1
    g1[3] = (int)((unsigned int)(TILE_ELEMS & 0xffff) << 16);  // tile_dim0
    g1[4] = (int)1u;                                           // tile_dim1=1, tile_dim2=0
    g1[5] = (int)(unsigned int)(t0 & 0xffffffffull);           // tensor_dim0_stride
    g1[6] = (int)(unsigned int)((t0 >> 32) & 0xffffull);
    g1[7] = 0;

    i32x4 z4; z4[0] = 0; z4[1] = 0; z4[2] = 0; z4[3] = 0;
    i32x8 z8; z8[0] = 0; z8[1] = 0; z8[2] = 0; z8[3] = 0;
    z8[4] = 0; z8[5] = 0; z8[6] = 0; z8[7] = 0;
    __builtin_amdgcn_tensor_load_to_lds(g0, g1, z4, z4, z8, 0);
    __builtin_amdgcn_s_wait_tensorcnt(0);
  }
  asm volatile("" ::: "memory");  // LDS written by TDM, invisible to the compiler
  __syncthreads();

  const long long row = (long long)blockIdx.x * TILE_ROWS + tid;
  float partial = 0.0f;
  if (row < BP) {
    const float* x = &sconf[tid * NC];
    float m = x[0];
#pragma unroll
    for (int j = 1; j < NC; ++j) m = fmaxf(m, x[j]);
    float s = 0.0f;
#pragma unroll
    for (int j = 0; j < NC; ++j) s += expf(x[j] - m);
    float lse = m + logf(s);
    int gt = conf_t[row];
    ce[row] = lse - x[gt];  // == per-prior CE == mining loss (before pos mask)
    if (gt > 0) {
#pragma unroll
      for (int j = 0; j < 4; ++j) {
        float d = loc_data[row * 4 + j] - loc_t[row * 4 + j];
        float ad = fabsf(d);
        partial += (ad < 1.0f) ? 0.5f * d * d : ad - 0.5f;
      }
    }
  }
  float wsum = wave_reduce_f32_wmma(partial);  // EXEC all-ones: no early exits above
  if ((tid & 31) == 0) atomicAdd(&acc[0], wsum);
}

// ------- Kernel 3: hard-negative mining via radix-select (one block per batch) -------
__global__ void k_mine(const float* __restrict__ ce,
                       const int* __restrict__ conf_t,
                       const int* __restrict__ num_pos,
                       float* __restrict__ acc /* [1]=loss_conf */) {
  const int b = blockIdx.x;
  const int tid = threadIdx.x;
  __shared__ unsigned int hist[256];
  __shared__ unsigned int s_pref, s_want, s_cntgt;
  __shared__ float s_sum;

  int np = num_pos[b];
  int k = NEGPOS * np; if (k > NP - 1) k = NP - 1;

  if (tid == 0) { s_sum = 0.0f; s_cntgt = 0; }
  __syncthreads();

  unsigned int tbits = 0xFFFFFFFFu;  // "nothing selected" sentinel (NaN bits, never matched)
  unsigned int rem = 0;
  if (k > 0) {
    unsigned int pref = 0, want = (unsigned int)k;
    for (int shift = 24; shift >= 0; shift -= 8) {
      hist[tid] = 0;
      __syncthreads();
      for (int p = tid; p < NP; p += 256) {
        bool pos = conf_t[b * NP + p] > 0;
        unsigned int v = pos ? 0u : __float_as_uint(ce[b * NP + p]);  // masked loss, >=0
        bool match = (shift == 24) || ((v >> (shift + 8)) == pref);
        if (match) atomicAdd(&hist[(v >> shift) & 255u], 1u);
      }
      __syncthreads();
      if (tid == 0) {
        unsigned int cum = 0; int chosen = 0; unsigned int nw = want;
        for (int bin = 255; bin >= 0; --bin) {
          if (cum + hist[bin] >= want) { chosen = bin; nw = want - cum; break; }
          cum += hist[bin];
        }
        s_pref = (pref << 8) | (unsigned int)chosen;
        s_want = nw;
      }
      __syncthreads();
      pref = s_pref; want = s_want;
      __syncthreads();
    }
    tbits = pref;  // k-th largest masked value (bit pattern)
    unsigned int lg = 0;
    for (int p = tid; p < NP; p += 256) {
      bool pos = conf_t[b * NP + p] > 0;
      unsigned int v = pos ? 0u : __float_as_uint(ce[b * NP + p]);
      if (v > tbits) ++lg;
    }
    atomicAdd(&s_cntgt, lg);
    __syncthreads();
    rem = (unsigned int)k - s_cntgt;  // #entries equal to threshold taken in index order
  }

  // positives + strictly-above-threshold negatives
  float lsum = 0.0f;
  for (int p = tid; p < NP; p += 256) {
    bool pos = conf_t[b * NP + p] > 0;
    float c = ce[b * NP + p];
    unsigned int v = pos ? 0u : __float_as_uint(c);
    if (pos || v > tbits) lsum += c;
  }
  atomicAdd(&s_sum, lsum);
  __syncthreads();
  if (tid == 0) {
    float extra = 0.0f;
    if (k > 0 && rem > 0) {  // stable-argsort tie-break: equals taken in ascending index
      unsigned int cnt = 0;
      for (int p = 0; p < NP && cnt < rem; ++p) {
        bool pos = conf_t[b * NP + p] > 0;
        unsigned int v = pos ? 0u : __float_as_uint(ce[b * NP + p]);
        if (v == tbits) { if (!pos) extra += ce[b * NP + p]; ++cnt; }
      }
    }
    atomicAdd(&acc[1], s_sum + extra);
  }
}

// ---------------- Kernel 4: finalize ----------------
__global__ void k_final(const float* __restrict__ acc,
                        const int* __restrict__ npos_total,
                        float* __restrict__ out) {
  if (threadIdx.x == 0 && blockIdx.x == 0) {
    float N = (float)(*npos_total);
    out[0] = acc[0] / N;
    out[1] = acc[1] / N;
  }
}

extern "C" void kernel_launch(void* const* d_in, const int* in_sizes, int n_in,
                              void* d_out, int out_size, void* d_ws, size_t ws_size,
                              hipStream_t stream) {
  const float* loc_data  = (const float*)d_in[0];
  const float* conf_data = (const float*)d_in[1];
  const float* priors    = (const float*)d_in[2];
  const float* targets   = (const float*)d_in[3];

  char* w = (char*)d_ws;
  size_t off = 0;
  float* bto    = (float*)(w + off); off += sizeof(float) * BP;
  int*   bti    = (int*)  (w + off); off += sizeof(int)   * BP;
  float* loc_t  = (float*)(w + off); off += sizeof(float) * BP * 4;
  int*   conf_t = (int*)  (w + off); off += sizeof(int)   * BP;
  float* cearr  = (float*)(w + off); off += sizeof(float) * BP;
  int*   numpos = (int*)  (w + off); off += 256;
  float* acc    = (float*)(w + off);           // [0]=loss_l, [1]=loss_conf
  int*   nptot  = (int*)(acc + 2);

  (void)hipMemsetAsync(acc, 0, 16, stream);  // zero accumulators (graph-capture safe)

  k_match<<<NB, 256, 0, stream>>>(priors, targets, bto, bti, loc_t, conf_t, numpos, nptot);
  k_confrow<<<BP / TILE_ROWS, 256, 0, stream>>>(conf_data, loc_data, loc_t, conf_t, cearr, acc);
  k_mine<<<NB, 256, 0, stream>>>(cearr, conf_t, numpos, acc);
  k_final<<<1, 32, 0, stream>>>(acc, nptot, (float*)d_out);
}